// ATSSAssigner_31576599560728
// MI455X (gfx1250) — compile-verified
//
#include <hip/hip_runtime.h>
#include <stdint.h>

// Problem constants (fixed by reference setup: 1280px image, strides 8/16/32)
#define LTOT   33600
#define NGT    100
#define NCLS   80
#define TOPKN  9
#define NLEV   3
#define CAND   (TOPKN*NLEV)

typedef float v2f __attribute__((ext_vector_type(2)));
typedef float v8f __attribute__((ext_vector_type(8)));

static __device__ __forceinline__ void insert9(unsigned long long (&a)[9],
                                               unsigned long long key) {
    if (key < a[8]) {
        a[8] = key;
#pragma unroll
        for (int q = 8; q > 0; --q) {
            unsigned long long x = a[q - 1], y = a[q];
            if (y < x) { a[q - 1] = y; a[q] = x; }
        }
    }
}

static __device__ __forceinline__ float iou_box(float g0, float g1, float g2, float g3,
                                                float areaG,
                                                float a0, float a1, float a2, float a3,
                                                float eps) {
    float ox = fminf(g2, a2) - fmaxf(g0, a0);
    float oy = fminf(g3, a3) - fmaxf(g1, a1);
    float ov = fmaxf(ox, 0.f) * fmaxf(oy, 0.f);
    float areaA = fmaxf(a2 - a0, 0.f) * fmaxf(a3 - a1, 0.f);
    return ov / (areaG + areaA - ov + eps);
}

// ---------------------------------------------------------------------------
// Kernel 0: zero the positive-mask region of the workspace
// ---------------------------------------------------------------------------
__global__ void atss_zero_kernel(uint32_t* __restrict__ p, int n32) {
    int i = blockIdx.x * blockDim.x + threadIdx.x;
    if (i < n32) p[i] = 0u;
}

// ---------------------------------------------------------------------------
// Kernel A: WMMA squared-distance tiles -> per-level top-9 -> ATSS threshold
//           -> scatter positive bytes into mask[B][NGT][LTOT].
// Grid: (ceil(NGT/16), B), 256 threads = 8 wave32.
// ---------------------------------------------------------------------------
__global__ __launch_bounds__(256)
void atss_topk_kernel(const float* __restrict__ anchors,     // [LTOT,4]
                      const float* __restrict__ gt_bboxes,   // [B,NGT,4]
                      const float* __restrict__ pad_mask,    // [B,NGT,1]
                      uint8_t* __restrict__ mask) {
    __shared__ float s_tile[16 * 128];                 // 16 gts x 128 anchors (d^2)
    __shared__ unsigned long long s_merge[16 * 16 * 9];
    __shared__ int   s_top[16][CAND];
    __shared__ float s_gb[16][4];
    __shared__ float s_gc[16][2];
    __shared__ float s_pad[16];

    const int b    = blockIdx.y;
    const int tile = blockIdx.x;          // group of 16 gts
    const int tid  = threadIdx.x;
    const int lane = tid & 31;
    const int wave = tid >> 5;            // 0..7
    const int sgt  = tid >> 4;            // gt-in-tile handled by this thread (0..15)
    const int slot = tid & 15;            // 16 threads cooperate per gt

    if (tid < 16) {
        int g = tile * 16 + tid;
        if (g < NGT) {
            const float* gb = gt_bboxes + ((size_t)b * NGT + g) * 4;
            float g0 = gb[0], g1 = gb[1], g2 = gb[2], g3 = gb[3];
            s_gb[tid][0] = g0; s_gb[tid][1] = g1; s_gb[tid][2] = g2; s_gb[tid][3] = g3;
            s_gc[tid][0] = (g0 + g2) * 0.5f;
            s_gc[tid][1] = (g1 + g3) * 0.5f;
            s_pad[tid]   = pad_mask[(size_t)b * NGT + g];
        } else {
            s_gb[tid][0] = s_gb[tid][1] = s_gb[tid][2] = s_gb[tid][3] = 0.f;
            s_gc[tid][0] = s_gc[tid][1] = 0.f;
            s_pad[tid] = 0.f;
        }
    }
    __syncthreads();

    // A fragment (16x4 f32): row m = [gx, gy, gx^2+gy^2, 1]; K0/K1 in lanes 0-15,
    // K2/K3 in lanes 16-31 (ISA 16x4 A layout).
    {
    }
    const int m = lane & 15;
    float gx = s_gc[m][0], gy = s_gc[m][1];
    v2f afrag;
    if (lane < 16) { afrag.x = gx; afrag.y = gy; }
    else           { afrag.x = gx * gx + gy * gy; afrag.y = 1.0f; }

    const int levStart[NLEV] = {0, 25600, 32000};
    const int levLen[NLEV]   = {25600, 6400, 1600};

    unsigned long long t9[9];

#pragma unroll
    for (int lev = 0; lev < NLEV; ++lev) {
        const int lBeg = levStart[lev];
        const int lEnd = levStart[lev] + levLen[lev];
#pragma unroll
        for (int q = 0; q < 9; ++q) t9[q] = ~0ULL;

        const int nChunks = (levLen[lev] + 127) >> 7;
        for (int ch = 0; ch < nChunks; ++ch) {
            const int base = lBeg + (ch << 7);
            // B fragment (4x16 f32): col n = [-2ax, -2ay, 1, ax^2+ay^2]
            int gidx = base + wave * 16 + (lane & 15);
            float ax = 3.0e8f, ay = 3.0e8f;   // sentinel: far away (d^2 finite)
            if (gidx < lEnd) {
                const float* ab = anchors + (size_t)gidx * 4;
                ax = (ab[0] + ab[2]) * 0.5f;
                ay = (ab[1] + ab[3]) * 0.5f;
            }
            v2f bfrag;
            if (lane < 16) { bfrag.x = -2.0f * ax; bfrag.y = -2.0f * ay; }
            else           { bfrag.x = 1.0f;       bfrag.y = ax * ax + ay * ay; }

            v8f c = {};
            // d^2(g,l) tile: D = A x B + 0
            c = __builtin_amdgcn_wmma_f32_16x16x4_f32(
                    false, afrag, false, bfrag, (short)0, c, false, false);

            const int mBase = (lane < 16) ? 0 : 8;
            const int col   = wave * 16 + (lane & 15);
#pragma unroll
            for (int r = 0; r < 8; ++r)
                s_tile[(mBase + r) * 128 + col] = c[r];
            __syncthreads();

            // 16 threads per gt scan 8 anchors each, keep sorted top-9 keys
#pragma unroll
            for (int k = 0; k < 8; ++k) {
                int j  = slot * 8 + k;
                int gi = base + j;
                if (gi < lEnd) {
                    float d = s_tile[sgt * 128 + j];
                    unsigned long long key =
                        (((unsigned long long)__float_as_uint(d)) << 32) |
                        (unsigned long long)(unsigned)gi;
                    insert9(t9, key);
                }
            }
            __syncthreads();
        }

        // merge the 16 partial top-9 lists per gt
#pragma unroll
        for (int q = 0; q < 9; ++q)
            s_merge[(sgt * 16 + slot) * 9 + q] = t9[q];
        __syncthreads();
        if (slot == 0) {
            unsigned long long f9[9];
#pragma unroll
            for (int q = 0; q < 9; ++q) f9[q] = ~0ULL;
            for (int t = 0; t < 16; ++t)
                for (int q = 0; q < 9; ++q)
                    insert9(f9, s_merge[(sgt * 16 + t) * 9 + q]);
#pragma unroll
            for (int q = 0; q < 9; ++q)
                s_top[sgt][lev * 9 + q] = (int)(f9[q] & 0xffffffffULL);
        }
        __syncthreads();
    }

    // ATSS threshold (mean + std over 27 candidate IoUs) and scatter positives
    if (slot == 0) {
        int g = tile * 16 + sgt;
        if (g < NGT && s_pad[sgt] > 0.f) {
            float g0 = s_gb[sgt][0], g1 = s_gb[sgt][1];
            float g2 = s_gb[sgt][2], g3 = s_gb[sgt][3];
            float areaG = fmaxf(g2 - g0, 0.f) * fmaxf(g3 - g1, 0.f);
            float iouv[CAND];
            float sum = 0.f, sumsq = 0.f;
            for (int q = 0; q < CAND; ++q) {
                int idx = s_top[sgt][q];
                const float* ab = anchors + (size_t)idx * 4;
                float v = iou_box(g0, g1, g2, g3, areaG,
                                  ab[0], ab[1], ab[2], ab[3], 1e-10f);
                iouv[q] = v; sum += v; sumsq += v * v;
            }
            float mean = sum * (1.0f / CAND);
            float var  = (sumsq - (float)CAND * mean * mean) * (1.0f / (CAND - 1));
            float thr  = mean + sqrtf(fmaxf(var, 0.f));
            uint8_t* mrow = mask + ((size_t)b * NGT + g) * LTOT;
            for (int q = 0; q < CAND; ++q) {
                if (iouv[q] > thr) {
                    int idx = s_top[sgt][q];
                    const float* ab = anchors + (size_t)idx * 4;
                    float acx = (ab[0] + ab[2]) * 0.5f;
                    float acy = (ab[1] + ab[3]) * 0.5f;
                    float mn = fminf(fminf(acx - g0, acy - g1),
                                     fminf(g2 - acx, g3 - acy));
                    if (mn > 1e-9f) mrow[idx] = 1;   // is_in_gts
                }
            }
        }
    }
}

// ---------------------------------------------------------------------------
// Kernel B: per-anchor resolution. One thread per (b,l).
// ---------------------------------------------------------------------------
__global__ __launch_bounds__(256)
void atss_resolve_kernel(const float* __restrict__ anchors,
                         const float* __restrict__ gt_bboxes,
                         const int*   __restrict__ gt_labels,
                         const float* __restrict__ pred,
                         const uint8_t* __restrict__ mask,
                         const int*   __restrict__ bgp,
                         float* __restrict__ out_labels,
                         float* __restrict__ out_bboxes,
                         int*   __restrict__ ws_lab,
                         float* __restrict__ ws_iou) {
    __shared__ float s_gb[NGT][4];
    __shared__ int   s_lab[NGT];
    const int b = blockIdx.y;
    const int l = blockIdx.x * blockDim.x + threadIdx.x;

    if (threadIdx.x < NGT) {
        const float* gb = gt_bboxes + ((size_t)b * NGT + threadIdx.x) * 4;
        s_gb[threadIdx.x][0] = gb[0]; s_gb[threadIdx.x][1] = gb[1];
        s_gb[threadIdx.x][2] = gb[2]; s_gb[threadIdx.x][3] = gb[3];
        s_lab[threadIdx.x] = gt_labels[(size_t)b * NGT + threadIdx.x];
    }
    __syncthreads();
    if (l >= LTOT) return;

    const float* ab = anchors + (size_t)l * 4;
    float a0 = ab[0], a1 = ab[1], a2 = ab[2], a3 = ab[3];
    float areaA = fmaxf(a2 - a0, 0.f) * fmaxf(a3 - a1, 0.f);

    const uint8_t* mcol = mask + (size_t)b * NGT * LTOT + l;
    int cnt = 0, firstg = -1, bestg = 0;
    float best = -1.0f;
    for (int g = 0; g < NGT; ++g) {
        int mg = mcol[(size_t)g * LTOT];
        cnt += mg;
        if (mg && firstg < 0) firstg = g;
        // ious[b,g,l] with eps 1e-10 (for is_max_iou argmax, first-max semantics)
        float v = iou_box(s_gb[g][0], s_gb[g][1], s_gb[g][2], s_gb[g][3],
                          fmaxf(s_gb[g][2] - s_gb[g][0], 0.f) *
                          fmaxf(s_gb[g][3] - s_gb[g][1], 0.f),
                          a0, a1, a2, a3, 1e-10f);
        (void)areaA;
        if (v > best) { best = v; bestg = g; }
    }

    const bool useMax = (cnt > 1);
    int agi, fcnt;
    if (useMax) { agi = bestg; fcnt = 1; }
    else        { agi = (cnt > 0) ? firstg : 0; fcnt = cnt; }

    const int bg  = *bgp;
    const int lab = (fcnt > 0) ? s_lab[agi] : bg;

    // pred_ious = max_g iou(gt_g, pred_l) * final_mask_g   (eps 1e-9)
    const float* pb = pred + ((size_t)b * LTOT + l) * 4;
    float p0 = pb[0], p1 = pb[1], p2 = pb[2], p3 = pb[3];
    float piou = 0.f;
    if (useMax) {
        float areaG = fmaxf(s_gb[bestg][2] - s_gb[bestg][0], 0.f) *
                      fmaxf(s_gb[bestg][3] - s_gb[bestg][1], 0.f);
        piou = fmaxf(iou_box(s_gb[bestg][0], s_gb[bestg][1],
                             s_gb[bestg][2], s_gb[bestg][3], areaG,
                             p0, p1, p2, p3, 1e-9f), 0.f);
    } else if (cnt > 0) {
        for (int g = 0; g < NGT; ++g) {
            if (mcol[(size_t)g * LTOT]) {
                float areaG = fmaxf(s_gb[g][2] - s_gb[g][0], 0.f) *
                              fmaxf(s_gb[g][3] - s_gb[g][1], 0.f);
                piou = fmaxf(piou, iou_box(s_gb[g][0], s_gb[g][1],
                                           s_gb[g][2], s_gb[g][3], areaG,
                                           p0, p1, p2, p3, 1e-9f));
            }
        }
    }

    const size_t o = (size_t)b * LTOT + l;
    out_labels[o] = (float)lab;
    float* ob = out_bboxes + o * 4;
    ob[0] = s_gb[agi][0]; ob[1] = s_gb[agi][1];
    ob[2] = s_gb[agi][2]; ob[3] = s_gb[agi][3];
    ws_lab[o] = lab;
    ws_iou[o] = piou;
}

// ---------------------------------------------------------------------------
// Kernel C: coalesced score expansion, one thread per (b,l,c)
// ---------------------------------------------------------------------------
__global__ void atss_scores_kernel(const int* __restrict__ ws_lab,
                                   const float* __restrict__ ws_iou,
                                   float* __restrict__ out_scores, int total) {
    int i = blockIdx.x * blockDim.x + threadIdx.x;
    if (i >= total) return;
    int c = i % NCLS;
    int a = i / NCLS;
    int lab = ws_lab[a];
    out_scores[i] = (lab == c) ? ws_iou[a] : 0.f;   // lab==bg(80) never matches c<80
}

// ---------------------------------------------------------------------------
extern "C" void kernel_launch(void* const* d_in, const int* in_sizes, int n_in,
                              void* d_out, int out_size, void* d_ws, size_t ws_size,
                              hipStream_t stream) {
    const float* anchors   = (const float*)d_in[0];
    const int*   gt_labels = (const int*)d_in[2];
    const float* gt_bboxes = (const float*)d_in[3];
    const float* pad       = (const float*)d_in[4];
    const int*   bgp       = (const int*)d_in[5];
    const float* pred      = (const float*)d_in[6];

    const int B = in_sizes[3] / (NGT * 4);   // 8

    // workspace layout: [mask u8 B*NGT*LTOT][lab i32 B*LTOT][piou f32 B*LTOT]
    uint8_t* mask = (uint8_t*)d_ws;
    size_t maskBytes = (size_t)B * NGT * LTOT;
    size_t off = (maskBytes + 255) & ~(size_t)255;
    int* ws_lab = (int*)((char*)d_ws + off);
    off += (((size_t)B * LTOT * sizeof(int)) + 255) & ~(size_t)255;
    float* ws_iou = (float*)((char*)d_ws + off);

    float* out        = (float*)d_out;
    float* out_labels = out;                            // [B,LTOT]
    float* out_bboxes = out + (size_t)B * LTOT;         // [B,LTOT,4]
    float* out_scores = out + (size_t)B * LTOT * 5;     // [B,LTOT,80]

    int n32 = (int)(maskBytes / 4);
    atss_zero_kernel<<<(n32 + 255) / 256, 256, 0, stream>>>((uint32_t*)mask, n32);

    dim3 gA((NGT + 15) / 16, B);
    atss_topk_kernel<<<gA, 256, 0, stream>>>(anchors, gt_bboxes, pad, mask);

    dim3 gB((LTOT + 255) / 256, B);
    atss_resolve_kernel<<<gB, 256, 0, stream>>>(anchors, gt_bboxes, gt_labels, pred,
                                                mask, bgp, out_labels, out_bboxes,
                                                ws_lab, ws_iou);

    int total = B * LTOT * NCLS;
    atss_scores_kernel<<<(total + 255) / 256, 256, 0, stream>>>(ws_lab, ws_iou,
                                                                out_scores, total);
}